// MaxwellModel_36155034698236
// MI455X (gfx1250) — compile-verified
//
#include <hip/hip_runtime.h>
#include <cstdint>

#define TLEN   4096   // time steps per row
#define WPB    8      // waves per block (wave32)
#define EPL    4      // elements per lane per tile
#define TILE   128    // 32 lanes * EPL
#define NTILES 32     // TLEN / TILE

__global__ __launch_bounds__(WPB * 32)
void maxwell_scan_kernel(const float* __restrict__ x, float* __restrict__ out, int rows)
{
    // 2 buffers * 8 waves * (32 lanes * 32B) = 16 KB LDS
    __shared__ float4 sbuf[2][WPB][64];

    const int lane = threadIdx.x & 31;
    const int w    = threadIdx.x >> 5;
    const int row  = blockIdx.x * WPB + w;
    if (row >= rows) return;

    const unsigned long long xbase = (unsigned long long)x;

    // global byte offset of this lane's first float2 of tile 0 (fits in u32: < 2^27)
    unsigned voff0 = ((unsigned)row * (unsigned)TLEN + (unsigned)(lane * EPL)) * 8u;

    // LDS byte addresses (low 32 bits of the generic pointer = LDS offset)
    unsigned laddr0 = (unsigned)(uintptr_t)&sbuf[0][w][lane * 2];
    unsigned laddr1 = (unsigned)(uintptr_t)&sbuf[1][w][lane * 2];

    // prologue: async-copy tile 0 into buffer 0 (32B per lane via 2x B128)
    asm volatile(
        "global_load_async_to_lds_b128 %0, %1, %2\n\t"
        "global_load_async_to_lds_b128 %0, %1, %2 offset:16"
        :: "v"(laddr0), "v"(voff0), "s"(xbase) : "memory");

    const float K = 2.0f;   // E / ETA
    float carry = 0.0f;     // gamma before first element of the row

    for (int i = 0; i < NTILES; ++i) {
        const int p = i & 1;

        if (i + 1 < NTILES) {
            unsigned voff = voff0 + (unsigned)(i + 1) * (TILE * 8u);
            unsigned ladr = (p == 0) ? laddr1 : laddr0;
            // ensure prior ds reads of the target buffer fully retired (WAR: ds vs async unordered)
            asm volatile("s_wait_dscnt 0" ::: "memory");
            asm volatile(
                "global_load_async_to_lds_b128 %0, %1, %2\n\t"
                "global_load_async_to_lds_b128 %0, %1, %2 offset:16"
                :: "v"(ladr), "v"(voff), "s"(xbase) : "memory");
            // async loads complete in order: <=2 outstanding means tile i has landed
            asm volatile("s_wait_asynccnt 2" ::: "memory");
        } else {
            asm volatile("s_wait_asynccnt 0" ::: "memory");
        }

        float4 q0 = sbuf[p][w][lane * 2];
        float4 q1 = sbuf[p][w][lane * 2 + 1];
        float e0 = q0.x, d0 = q0.y, e1 = q0.z, d1 = q0.w;
        float e2 = q1.x, d2 = q1.y, e3 = q1.z, d3 = q1.w;

        // per-element affine terms: gamma = a*gamma_prev + b
        float a0 = fmaf(-K, d0, 1.0f), b0 = K * d0 * e0;
        float a1 = fmaf(-K, d1, 1.0f), b1 = K * d1 * e1;
        float a2 = fmaf(-K, d2, 1.0f), b2 = K * d2 * e2;
        float a3 = fmaf(-K, d3, 1.0f), b3 = K * d3 * e3;

        // lane-local inclusive composition over 4 elements
        float A1 = a1 * a0, B1 = fmaf(a1, b0, b1);
        float A2 = a2 * A1, B2 = fmaf(a2, B1, b2);
        float A3 = a3 * A2, B3 = fmaf(a3, B2, b3);

        // wave32 inclusive scan of lane totals under (a2*a1, a2*b1+b2)
        float sa = A3, sb = B3;
        #pragma unroll
        for (int d = 1; d < 32; d <<= 1) {
            float ua = __shfl_up(sa, d, 32);
            float ub = __shfl_up(sb, d, 32);
            if (lane >= d) { sb = fmaf(sa, ub, sb); sa *= ua; }
        }

        // exclusive prefix for this lane (identity for lane 0)
        float ea = __shfl_up(sa, 1, 32);
        float eb = __shfl_up(sb, 1, 32);
        if (lane == 0) { ea = 1.0f; eb = 0.0f; }

        float g_in = fmaf(ea, carry, eb);          // gamma entering this lane's chunk
        float g0 = fmaf(a0, g_in, b0);
        float g1 = fmaf(A1, g_in, B1);
        float g2 = fmaf(A2, g_in, B2);
        float g3 = fmaf(A3, g_in, B3);

        carry = __shfl(g3, 31, 32);                // gamma at end of tile

        // sigma = 0.5*eps + 2*(eps - gamma) = 2.5*eps - 2*gamma
        float4 o;
        o.x = fmaf(-2.0f, g0, 2.5f * e0);
        o.y = fmaf(-2.0f, g1, 2.5f * e1);
        o.z = fmaf(-2.0f, g2, 2.5f * e2);
        o.w = fmaf(-2.0f, g3, 2.5f * e3);

        size_t oidx = (size_t)row * TLEN + (size_t)i * TILE + (size_t)(lane * EPL);
        *(float4*)(out + oidx) = o;
    }
}

extern "C" void kernel_launch(void* const* d_in, const int* in_sizes, int n_in,
                              void* d_out, int out_size, void* d_ws, size_t ws_size,
                              hipStream_t stream) {
    const float* x = (const float*)d_in[0];
    float* out = (float*)d_out;
    (void)n_in; (void)d_ws; (void)ws_size; (void)out_size;

    long long n = (long long)in_sizes[0];        // B * T * 2
    int rows = (int)(n / (2LL * TLEN));          // B

    dim3 block(WPB * 32);
    dim3 grid((unsigned)((rows + WPB - 1) / WPB));
    maxwell_scan_kernel<<<grid, block, 0, stream>>>(x, out, rows);
}